// SpiralConvConvBlock_45518063403509
// MI455X (gfx1250) — compile-verified
//
#include <hip/hip_runtime.h>

// Problem constants (from reference): x is [L, B, D] float32.
#define L_SEQ  4096
#define BATCH  4
#define DIM    1024
#define BD     (BATCH * DIM)        // 4096 channels (b*D + d)
#define NCHUNK 64                   // chunks along L
#define TCHUNK (L_SEQ / NCHUNK)     // 64 rows per chunk
#define DEPTH  8                    // async prefetch depth (rows in flight)
#define TPB    256                  // 8 waves of 32

// ---------------- CDNA5 async global->LDS path (guarded) ----------------
#if __has_builtin(__builtin_amdgcn_global_load_async_to_lds_b128)
#define HAVE_ASYNC 1
#else
#define HAVE_ASYNC 0
#endif

#if __has_builtin(__builtin_amdgcn_s_wait_asynccnt)
#define WAIT_ASYNC(n) __builtin_amdgcn_s_wait_asynccnt(n)
#elif defined(__gfx1250__)
#define WAIT_ASYNC(n) asm volatile("s_wait_asynccnt %0" ::"i"(n) : "memory")
#else
#define WAIT_ASYNC(n) ((void)0)
#endif

typedef int v4i_t __attribute__((ext_vector_type(4)));
typedef __attribute__((address_space(1))) v4i_t* gv4_ptr;   // global v4i32*
typedef __attribute__((address_space(3))) v4i_t* lv4_ptr;   // LDS    v4i32*

// Each lane moves its own 16B slice global -> LDS (ASYNCcnt-tracked on CDNA5).
__device__ __forceinline__ void async_row_load(const float* g, float* l) {
#if HAVE_ASYNC
  __builtin_amdgcn_global_load_async_to_lds_b128((gv4_ptr)g, (lv4_ptr)l, 0, 0);
#else
  *(float4*)l = *(const float4*)g;   // synchronous fallback
#endif
}

// Per-lane coefficients for 4 consecutive channels d0..d0+3:
// c = gamma * e^{i theta}, gamma = exp(-exp(lmlg)).
// Arguments are N(0,1) draws -> hardware transcendentals are accurate here.
__device__ __forceinline__ void channel_coeffs(const float* __restrict__ lmlg,
                                               const float* __restrict__ theta,
                                               int d0, float cr[4], float ci[4]) {
  float4 lm = *(const float4*)(lmlg + d0);
  float4 th = *(const float4*)(theta + d0);
  float lmv[4] = {lm.x, lm.y, lm.z, lm.w};
  float thv[4] = {th.x, th.y, th.z, th.w};
#pragma unroll
  for (int j = 0; j < 4; ++j) {
    float g = __expf(-__expf(lmv[j]));
    float s, c;
    __sincosf(thv[j], &s, &c);
    cr[j] = g * c;
    ci[j] = g * s;
  }
}

// ---------------- Pass 1: chunk-local scan, zero init; emit end states ----------------
__global__ __launch_bounds__(TPB) void spiral_pass1(const float* __restrict__ x,
                                                    const float* __restrict__ lmlg,
                                                    const float* __restrict__ theta,
                                                    float* __restrict__ cc) {
  __shared__ float4 buf[DEPTH][TPB];          // 32 KB
  const int tid = threadIdx.x;
  const int b   = blockIdx.x;                 // batch tile (0..3)
  const int k   = blockIdx.y;                 // chunk (0..63)
  const int d0  = tid * 4;                    // 4 consecutive channels per lane

  float cr[4], ci[4];
  channel_coeffs(lmlg, theta, d0, cr, ci);

  const float* xp = x + (size_t)k * TCHUNK * BD + (size_t)b * DIM + d0;

#pragma unroll
  for (int i = 0; i < DEPTH; ++i)
    async_row_load(xp + (size_t)i * BD, (float*)&buf[i][tid]);

  float sr[4] = {0.f, 0.f, 0.f, 0.f};
  float si[4] = {0.f, 0.f, 0.f, 0.f};

#pragma unroll DEPTH
  for (int t = 0; t < TCHUNK; ++t) {
    WAIT_ASYNC(DEPTH - 1);                    // oldest row resident in LDS
    float4 xv = buf[t & (DEPTH - 1)][tid];
    float xs[4] = {xv.x, xv.y, xv.z, xv.w};
#pragma unroll
    for (int j = 0; j < 4; ++j) {             // s = c*s + x
      float nr = fmaf(cr[j], sr[j], fmaf(-ci[j], si[j], xs[j]));
      float ni = fmaf(ci[j], sr[j], cr[j] * si[j]);
      sr[j] = nr;
      si[j] = ni;
    }
    if (t + DEPTH < TCHUNK)                   // reissue into the slot just consumed
      async_row_load(xp + (size_t)(t + DEPTH) * BD, (float*)&buf[t & (DEPTH - 1)][tid]);
  }

  // store chunk-local end states as interleaved (re, im)
  float* cp = cc + 2 * ((size_t)k * BD + (size_t)b * DIM + d0);
  *(float4*)(cp)     = make_float4(sr[0], si[0], sr[1], si[1]);
  *(float4*)(cp + 4) = make_float4(sr[2], si[2], sr[3], si[3]);
}

// ---------------- Pass 3: seeded re-scan, write out = Re(s) * x ----------------
__global__ __launch_bounds__(TPB) void spiral_pass3(const float* __restrict__ x,
                                                    const float* __restrict__ lmlg,
                                                    const float* __restrict__ theta,
                                                    const float* __restrict__ cc,
                                                    float* __restrict__ out) {
  __shared__ float4 buf[DEPTH][TPB];
  const int tid = threadIdx.x;
  const int b   = blockIdx.x;
  const int k   = blockIdx.y;
  const int d0  = tid * 4;

  float cr[4], ci[4];
  channel_coeffs(lmlg, theta, d0, cr, ci);

  const float* cp = cc + 2 * ((size_t)k * BD + (size_t)b * DIM + d0);
  float4 c01 = *(const float4*)cp;
  float4 c23 = *(const float4*)(cp + 4);
  float sr[4] = {c01.x, c01.z, c23.x, c23.z};
  float si[4] = {c01.y, c01.w, c23.y, c23.w};

  const float* xp = x + (size_t)k * TCHUNK * BD + (size_t)b * DIM + d0;
  float*       op = out + (size_t)k * TCHUNK * BD + (size_t)b * DIM + d0;

#pragma unroll
  for (int i = 0; i < DEPTH; ++i)
    async_row_load(xp + (size_t)i * BD, (float*)&buf[i][tid]);

#pragma unroll DEPTH
  for (int t = 0; t < TCHUNK; ++t) {
    WAIT_ASYNC(DEPTH - 1);
    float4 xv = buf[t & (DEPTH - 1)][tid];
    float xs[4] = {xv.x, xv.y, xv.z, xv.w};
#pragma unroll
    for (int j = 0; j < 4; ++j) {
      float nr = fmaf(cr[j], sr[j], fmaf(-ci[j], si[j], xs[j]));
      float ni = fmaf(ci[j], sr[j], cr[j] * si[j]);
      sr[j] = nr;
      si[j] = ni;
    }
    if (t + DEPTH < TCHUNK)
      async_row_load(xp + (size_t)(t + DEPTH) * BD, (float*)&buf[t & (DEPTH - 1)][tid]);
    *(float4*)(op + (size_t)t * BD) =
        make_float4(sr[0] * xs[0], sr[1] * xs[1], sr[2] * xs[2], sr[3] * xs[3]);
  }
}

// ---------------- Pass 2: serial carry combine per channel (in place) ----------------
// carry_in[0] = (last_conv_init[d], 0); carry_in[k] = c^T * carry_in[k-1] + end[k-1]
// Uses PRECISE sincosf/expf: the argument is T*theta (up to ~ +-300), where the
// hardware fast path loses accuracy. This kernel is 16 blocks; cost irrelevant.
__global__ __launch_bounds__(TPB) void spiral_pass2(const float* __restrict__ lmlg,
                                                    const float* __restrict__ theta,
                                                    const float* __restrict__ lci,
                                                    float* __restrict__ cc) {
  const int ch = blockIdx.x * TPB + threadIdx.x;  // 0..4095
  const int d  = ch & (DIM - 1);

  // closed form: c^T = exp(-T*exp(lmlg)) * e^{i*T*theta}
  float gT = expf(-(float)TCHUNK * expf(lmlg[d]));
  float sa, ca;
  sincosf((float)TCHUNK * theta[d], &sa, &ca);
  const float cTr = gT * ca, cTi = gT * sa;

  float car_r = lci[d], car_i = 0.0f;
  for (int kk = 0; kk < NCHUNK; ++kk) {
    size_t idx = 2 * ((size_t)kk * BD + ch);
    float er = cc[idx], ei = cc[idx + 1];   // chunk-local end state
    cc[idx]     = car_r;                    // overwrite with carry-in for this chunk
    cc[idx + 1] = car_i;
    float nr = fmaf(cTr, car_r, fmaf(-cTi, car_i, er));
    float ni = fmaf(cTi, car_r, fmaf(cTr, car_i, ei));
    car_r = nr;
    car_i = ni;
  }
}

extern "C" void kernel_launch(void* const* d_in, const int* in_sizes, int n_in,
                              void* d_out, int out_size, void* d_ws, size_t ws_size,
                              hipStream_t stream) {
  const float* x     = (const float*)d_in[0];  // [L,B,D]
  const float* lmlg  = (const float*)d_in[1];  // [D]
  const float* theta = (const float*)d_in[2];  // [D]
  const float* lci   = (const float*)d_in[3];  // [D]
  float*       out   = (float*)d_out;          // [L,B,D]
  float*       cc    = (float*)d_ws;           // 2*NCHUNK*BD floats = 2 MB scratch

  dim3 grid1(BATCH, NCHUNK);
  spiral_pass1<<<grid1, TPB, 0, stream>>>(x, lmlg, theta, cc);
  spiral_pass2<<<BD / TPB, TPB, 0, stream>>>(lmlg, theta, lci, cc);
  spiral_pass3<<<grid1, TPB, 0, stream>>>(x, lmlg, theta, cc, out);
}